// OrthogonalLinear_56813827392404
// MI455X (gfx1250) — compile-verified
//
#include <hip/hip_runtime.h>

#define D   512
#define BSZ 32768
#define NB  8
#define BLK 64

typedef float v2f __attribute__((ext_vector_type(2)));
typedef float v8f __attribute__((ext_vector_type(8)));

// ---------------------------------------------------------------------------
// K1: S_all[b][i][j] = <u_{64b+i}, u_{64b+j}> with u_r[t] = U[r][t]*(t>=r)
// ---------------------------------------------------------------------------
__global__ void k_S(const float* __restrict__ U, float* __restrict__ S_all) {
  int b = blockIdx.x;            // 0..7
  int i = blockIdx.y;            // 0..63
  int j = threadIdx.x;           // 0..63
  int ra = b * BLK + i, rb = b * BLK + j;
  int lo = ra > rb ? ra : rb;
  const float* Ua = U + ra * D;
  const float* Ub = U + rb * D;
  float s = 0.f;
  for (int t = lo; t < D; ++t) s += Ua[t] * Ub[t];
  S_all[(b * BLK + i) * BLK + j] = s;
}

// ---------------------------------------------------------------------------
// K2: per-block T recurrence (LAPACK larft-style), T upper triangular 64x64.
// T[k,k]=tau_k ; T[0:k,k] = -tau_k * T[0:k,0:k] @ S[0:k,k]
// One 64-thread workgroup per block; S and T live in LDS.
// ---------------------------------------------------------------------------
__global__ void k_T(const float* __restrict__ S_all, float* __restrict__ T_all) {
  __shared__ float S[BLK][BLK];
  __shared__ float T[BLK][BLK];
  int b = blockIdx.x;
  int r = threadIdx.x;           // 0..63
  for (int c = 0; c < BLK; ++c) {
    S[c][r] = S_all[(b * BLK + c) * BLK + r];
    T[c][r] = 0.f;
  }
  __syncthreads();
  for (int k = 0; k < BLK; ++k) {
    float tau = 2.0f / S[k][k];
    if (r == k) {
      T[k][k] = tau;
    } else if (r < k) {
      float w = 0.f;
      for (int c = r; c < k; ++c) w += T[r][c] * S[c][k];
      T[r][k] = -tau * w;
    }
    __syncthreads();
  }
  for (int c = 0; c < BLK; ++c) T_all[(b * BLK + c) * BLK + r] = T[c][r];
}

// ---------------------------------------------------------------------------
// K3: W_all[b][j][k] = sum_i V_b[j,i] * T_b[i,k],  V_b[j,i]=U[64b+i][j]*(j>=64b+i)
// One block per (b,j); 64 threads over k.
// ---------------------------------------------------------------------------
__global__ void k_W(const float* __restrict__ U, const float* __restrict__ T_all,
                    float* __restrict__ W_all) {
  int b = blockIdx.x;
  int j = blockIdx.y;
  int k = threadIdx.x;           // 0..63
  __shared__ float u[BLK];
  int row = b * BLK + k;
  u[k] = (j >= row) ? U[row * D + j] : 0.f;
  __syncthreads();
  float s = 0.f;
  for (int i = 0; i <= k; ++i)   // T is upper triangular
    s += u[i] * T_all[(b * BLK + i) * BLK + k];
  W_all[(b * D + j) * BLK + k] = s;
}

// ---------------------------------------------------------------------------
// K4: Q = I
// ---------------------------------------------------------------------------
__global__ void k_qinit(float* __restrict__ Q) {
  int idx = blockIdx.x * blockDim.x + threadIdx.x;
  Q[idx] = ((idx / D) == (idx % D)) ? 1.f : 0.f;
}

// ---------------------------------------------------------------------------
// K5a (per stage b): P[r][k] = sum_{j>=64b} Q[r][j] * W_b[j][k]
// ---------------------------------------------------------------------------
__global__ void k_P(const float* __restrict__ Q, const float* __restrict__ W_all,
                    float* __restrict__ P, int b) {
  int k = threadIdx.x;                                // 0..63
  int r = blockIdx.x * blockDim.y + threadIdx.y;      // 0..511
  float s = 0.f;
  for (int j = b * BLK; j < D; ++j)
    s += Q[r * D + j] * W_all[(b * D + j) * BLK + k];
  P[r * BLK + k] = s;
}

// ---------------------------------------------------------------------------
// K5b (per stage b): Q[r][j] -= sum_k P[r][k] * U[64b+k][j]*(j>=64b+k)
// ---------------------------------------------------------------------------
__global__ void k_upd(float* __restrict__ Q, const float* __restrict__ P,
                      const float* __restrict__ U, int b) {
  int j = blockIdx.x * 64 + threadIdx.x;
  int r = blockIdx.y * blockDim.y + threadIdx.y;
  int klim = j - b * BLK;
  if (klim > BLK - 1) klim = BLK - 1;
  if (klim < 0) return;
  float s = 0.f;
  for (int k = 0; k <= klim; ++k)
    s += P[r * BLK + k] * U[(b * BLK + k) * D + j];
  Q[r * D + j] -= s;
}

// ---------------------------------------------------------------------------
// K6: QT[n][k] = Q[k][n]  (so B fragments of the WMMA GEMM are contiguous b64)
// ---------------------------------------------------------------------------
__global__ void k_tr(const float* __restrict__ Q, float* __restrict__ QT) {
  int idx = blockIdx.x * blockDim.x + threadIdx.x;
  int n = idx / D, k = idx % D;
  QT[idx] = Q[k * D + n];
}

// ---------------------------------------------------------------------------
// K7: out[m][n] = sum_k x[m][k]*Q[k][n] + bias[n], via V_WMMA_F32_16X16X4_F32.
// Block: 256 threads = 8 waves in a 4(M) x 2(N) arrangement.
// Wave (mw,nw): rows [blk*64+mw*16, +16), cols [nw*256, +256) -> 16 C tiles.
// A frag (16x4 f32, 2 VGPR): lane L holds x[m0+L%16][k0+2*(L/16) .. +1]
// B frag (4x16 f32, 2 VGPR): lane L holds Q[k0+2*(L/16)..+1][n0+L%16] = QT rows
// C/D (16x16 f32, 8 VGPR):   VGPR g, lane L -> M=g+8*(L/16), N=L%16
// __launch_bounds__(256,1): allow the full VGPR budget so the 16 accumulator
// tiles (128 VGPRs) stay resident — round 1 showed them spilling to scratch.
// ---------------------------------------------------------------------------
__global__ void __launch_bounds__(256, 1)
k_gemm(const float* __restrict__ x, const float* __restrict__ QT,
       const float* __restrict__ bias, float* __restrict__ out) {
  int lane = threadIdx.x & 31;
  int w    = threadIdx.x >> 5;      // 0..7
  int mw   = w & 3;                 // 0..3
  int nw   = w >> 2;                // 0..1
  int m0   = blockIdx.x * 64 + mw * 16;
  int n0   = nw * 256;
  int lm   = lane & 15;
  int lk   = (lane >> 4) * 2;       // 0 or 2

  const float* xrow  = x  + (m0 + lm) * D + lk;   // A fragment base
  const float* qbase = QT + (n0 + lm) * D + lk;   // B fragment base (tile 0)

  v8f zero = {0.f, 0.f, 0.f, 0.f, 0.f, 0.f, 0.f, 0.f};
  v8f acc[16];
#pragma unroll
  for (int t = 0; t < 16; ++t) acc[t] = zero;

  for (int k0 = 0; k0 < D; k0 += 4) {
    v2f a = *(const v2f*)(xrow + k0);
#pragma unroll
    for (int t = 0; t < 16; ++t) {
      // tile t shifts N by t*16 -> QT rows advance by t*16*D (folds into the
      // 24-bit instruction offset: t * 32768 bytes)
      v2f bfr = *(const v2f*)(qbase + t * 16 * D + k0);
      acc[t] = __builtin_amdgcn_wmma_f32_16x16x4_f32(
          /*neg_a=*/false, a, /*neg_b=*/false, bfr,
          /*c_mod=*/(short)0, acc[t], /*reuse_a=*/false, /*reuse_b=*/false);
    }
  }

#pragma unroll
  for (int t = 0; t < 16; ++t) {
    int n = n0 + t * 16 + lm;
    float bv = bias[n];
#pragma unroll
    for (int g = 0; g < 8; ++g) {
      int m = m0 + g + 8 * (lane >> 4);
      // streamed once, never re-read: non-temporal store keeps L2 for QT
      __builtin_nontemporal_store(acc[t][g] + bv, &out[m * D + n]);
    }
  }
}

// ---------------------------------------------------------------------------
extern "C" void kernel_launch(void* const* d_in, const int* in_sizes, int n_in,
                              void* d_out, int out_size, void* d_ws, size_t ws_size,
                              hipStream_t stream) {
  const float* x    = (const float*)d_in[0];   // [32768, 512]
  const float* U    = (const float*)d_in[1];   // [512, 512]
  const float* bias = (const float*)d_in[2];   // [1, 512]
  float* out = (float*)d_out;

  float* ws = (float*)d_ws;
  float* Q   = ws;                       // 512*512
  float* QT  = Q  + D * D;               // 512*512
  float* S   = QT + D * D;               // 8*64*64
  float* T   = S  + NB * BLK * BLK;      // 8*64*64
  float* W   = T  + NB * BLK * BLK;      // 8*512*64
  float* P   = W  + NB * D * BLK;        // 512*64

  // Block Gram matrices (parallel)
  k_S<<<dim3(NB, BLK), BLK, 0, stream>>>(U, S);
  // Per-block triangular T factors (only serial part: 64 barrier steps)
  k_T<<<NB, BLK, 0, stream>>>(S, T);
  // W_b = V_b T_b (parallel over all blocks)
  k_W<<<dim3(NB, D), BLK, 0, stream>>>(U, T, W);
  // Q = I, then fold in the 8 block reflectors: Q <- Q - (Q W_b) V_b^T
  k_qinit<<<(D * D) / 256, 256, 0, stream>>>(Q);
  for (int b = 0; b < NB; ++b) {
    k_P<<<D / 4, dim3(BLK, 4), 0, stream>>>(Q, W, P, b);
    k_upd<<<dim3(D / 64, D / 4), dim3(64, 4), 0, stream>>>(Q, P, U, b);
  }
  // Transpose for contiguous WMMA B fragments
  k_tr<<<(D * D) / 256, 256, 0, stream>>>(Q, QT);
  // The big memory-bound GEMM: out = x @ Q + bias
  k_gemm<<<BSZ / 64, 256, 0, stream>>>(x, QT, bias, out);
}